// GAT_32641751449828
// MI455X (gfx1250) — compile-verified
//
#include <hip/hip_runtime.h>
#include <hip/hip_bf16.h>
#include <math.h>
#include <stdint.h>

typedef __attribute__((ext_vector_type(2))) float v2f;
typedef __attribute__((ext_vector_type(8))) float v8f;
typedef __attribute__((ext_vector_type(4))) unsigned int u32x4;
typedef __attribute__((ext_vector_type(8))) int i32x8;
typedef __attribute__((ext_vector_type(4))) int i32x4;

#define WG 256
#define TILE_ROWS 128
#define KCHUNK 32
#define ASTRIDE 36   // 32 data DWORDs + 4 pad DWORDs per row: bank = (36r+c)%64 conflict-free

#if defined(__HIP_DEVICE_COMPILE__) && \
    __has_builtin(__builtin_amdgcn_tensor_load_to_lds) && \
    __has_builtin(__builtin_amdgcn_s_wait_tensorcnt)
#define USE_TDM 1
#else
#define USE_TDM 0
#endif

__device__ __forceinline__ void atomic_add_f32(float* p, float v) {
    __hip_atomic_fetch_add(p, v, __ATOMIC_RELAXED, __HIP_MEMORY_SCOPE_AGENT);
}

// float atomic max: int-max for v>=0, uint-min for v<0; init must be -inf.
__device__ __forceinline__ void atomic_max_f32(float* p, float v) {
    if (v >= 0.0f) atomicMax((int*)p, __float_as_int(v));
    else           atomicMin((unsigned int*)p, __float_as_uint(v));
}

#if USE_TDM
// 2D tile load Global->LDS via Tensor Data Mover (ISA cdna5 ch.7/8).
// tensor_d0/d1 are the remaining valid extents measured from the tile start
// (elements); reads beyond them return zero. d0_stride in elements.
// pad_interval_code: pad after 2<<code DWORDs; pad_amount_code: (code+1) DWORDs.
__device__ __forceinline__ void tdm_load_2d(unsigned lds_addr, const void* gptr,
                                            unsigned tensor_d0, unsigned tensor_d1,
                                            unsigned tile_d0, unsigned tile_d1,
                                            unsigned long long d0_stride_elems,
                                            unsigned pad_interval_code,
                                            unsigned pad_amount_code,
                                            bool pad_en) {
    unsigned long long ga = (unsigned long long)(uintptr_t)gptr;
    u32x4 g0;
    g0[0] = 1u;                                             // count=1, load, user mode
    g0[1] = lds_addr;                                       // LDS byte address
    g0[2] = (unsigned)(ga & 0xffffffffull);                 // global_addr[31:0]
    g0[3] = (unsigned)((ga >> 32) & 0x1ffffffull)           // global_addr[56:32]
          | (2u << 30);                                     // type = 2 ("image")
    i32x8 g1;
    g1[0] = (int)((2u << 16) |                              // data_size = 4 bytes
                  (pad_en ? (1u << 20) : 0u) |              // pad_enable
                  (pad_interval_code << 22) |
                  (pad_amount_code << 25));                  // wg_mask=0, no atomic barrier
    g1[1] = (int)((tensor_d0 & 0xffffu) << 16);             // tensor_dim0[15:0]
    g1[2] = (int)(((tensor_d0 >> 16) & 0xffffu) |
                  ((tensor_d1 & 0xffffu) << 16));           // dim0[31:16] | dim1[15:0]
    g1[3] = (int)(((tensor_d1 >> 16) & 0xffffu) |
                  ((tile_d0 & 0xffffu) << 16));             // dim1[31:16] | tile_dim0
    g1[4] = (int)(tile_d1 & 0xffffu);                       // tile_dim1; tile_dim2 = 0
    g1[5] = (int)(d0_stride_elems & 0xffffffffull);         // tensor_dim0_stride[31:0]
    g1[6] = (int)((d0_stride_elems >> 32) & 0xffffull);     // stride[47:32]; dim1_stride=0
    g1[7] = 0;
    i32x4 gz4 = {0, 0, 0, 0};                               // 2D: groups 2/3 unused
    i32x8 gz8 = {0, 0, 0, 0, 0, 0, 0, 0};                   // clang-23 6-arg form extra group
    __builtin_amdgcn_tensor_load_to_lds(g0, g1, gz4, gz4, gz8, 0);
}
#endif

__global__ void fill_kernel(float* __restrict__ p, float v, long long n) {
    long long t = (long long)blockIdx.x * blockDim.x + threadIdx.x;
    if (t < n) p[t] = v;
}

// C[M,NC] = A[M,K] @ B[K,NC], row-major, fp32 WMMA 16x16x4.
// grid = (ceil(M/128), NC/16), block = 256 (8 wave32).
// Wave w computes rows [rowBase+16w, +16), cols [colBase, +16).
// Tiles staged into LDS by the Tensor Data Mover (wave 0 issues, TENSORcnt sync).
__global__ __launch_bounds__(WG)
void gemm_wmma_f32(const float* __restrict__ A, const float* __restrict__ B,
                   float* __restrict__ C, int M, int K, int NC) {
    __shared__ float Bs[512 * 16];              // full B column panel (K <= 512)
    __shared__ float As[TILE_ROWS * ASTRIDE];   // 128 x 32 A chunk, padded to 36

    const int tid  = threadIdx.x;
    const int wave = tid >> 5;
    const int lane = tid & 31;
    const int lmod = lane & 15;   // N (B/C) or M (A) index within tile
    const int lh   = lane >> 4;   // half-wave selector
    const int rowBase = blockIdx.x * TILE_ROWS;
    const int colBase = blockIdx.y * 16;

#if USE_TDM
    if (tid < 32) {   // one wave issues the DMA; completion covered by first tensorcnt wait
        tdm_load_2d((unsigned)(uintptr_t)Bs, B + colBase,
                    (unsigned)(NC - colBase), (unsigned)K,
                    16u, (unsigned)K,
                    (unsigned long long)NC, 0u, 0u, false);
    }
#else
    for (int idx = tid; idx < K * 16; idx += WG)
        Bs[idx] = B[(idx >> 4) * NC + colBase + (idx & 15)];
#endif

    v8f acc = {0.f, 0.f, 0.f, 0.f, 0.f, 0.f, 0.f, 0.f};

    for (int kc = 0; kc < K; kc += KCHUNK) {
#if USE_TDM
        if (tid < 32) {
            // 128x32 tile; rows past (M - rowBase) zero-filled by TDM OOB handling.
            // Pad 4 DWORDs after every 32 DWORDs -> LDS row stride 36 (ASTRIDE).
            tdm_load_2d((unsigned)(uintptr_t)As,
                        A + (size_t)rowBase * K + kc,
                        (unsigned)(K - kc), (unsigned)(M - rowBase),
                        (unsigned)KCHUNK, (unsigned)TILE_ROWS,
                        (unsigned long long)K,
                        4u /* pad after 32 DWORDs */, 3u /* 4 DWORDs */, true);
            __builtin_amdgcn_s_wait_tensorcnt(0);
        }
        __syncthreads();   // LDS tile visible to all waves
#else
        __syncthreads();   // prior reads done before overwrite
        for (int idx = tid; idx < TILE_ROWS * KCHUNK; idx += WG) {
            int r = idx >> 5;          // KCHUNK == 32
            int c = idx & 31;
            int row = rowBase + r; if (row >= M) row = M - 1;
            As[r * ASTRIDE + c] = A[(long long)row * K + kc + c];
        }
        __syncthreads();
#endif

        // A frag: lanes 0-15 hold K = kk,kk+1 ; lanes 16-31 hold K = kk+2,kk+3
        const float* arow = &As[(16 * wave + lmod) * ASTRIDE + 2 * lh];
#pragma unroll
        for (int kk = 0; kk < KCHUNK; kk += 4) {
            v2f a, b;
            a[0] = arow[kk + 0];
            a[1] = arow[kk + 1];
            // B frag: VGPR0 = row kk (+2 for hi half), VGPR1 = row kk+1 (+2)
            b[0] = Bs[(kk + 2 * lh + 0) * 16 + lmod];
            b[1] = Bs[(kk + 2 * lh + 1) * 16 + lmod];
            acc = __builtin_amdgcn_wmma_f32_16x16x4_f32(
                false, a, false, b, (short)0, acc, false, false);
        }
#if USE_TDM
        __syncthreads();   // all waves done reading As before next DMA overwrites it
#endif
    }

    // C/D layout: VGPR i -> M = i (lanes 0-15), M = i+8 (lanes 16-31); N = lmod
#pragma unroll
    for (int i = 0; i < 8; i++) {
        int row = rowBase + 16 * wave + 8 * lh + i;
        if (row < M) C[(long long)row * NC + colBase + lmod] = acc[i];
    }
}

// alpha_{s,d}[n,h] = sum_c h[n,h,c] * a_{src,dst}[h,c]
__global__ void alpha_kernel(const float* __restrict__ h, const float* __restrict__ a_src,
                             const float* __restrict__ a_dst, int N, int H, int Cc,
                             float* __restrict__ as, float* __restrict__ ad) {
    int t = blockIdx.x * blockDim.x + threadIdx.x;
    if (t >= N * H) return;
    int hh = t % H;
    const float* hp = h + (long long)t * Cc;   // h[n, hh, :]
    float ss = 0.f, dd = 0.f;
    for (int c = 0; c < Cc; c++) {
        float hv = hp[c];
        ss += hv * a_src[hh * Cc + c];
        dd += hv * a_dst[hh * Cc + c];
    }
    as[t] = ss; ad[t] = dd;
}

// Pass 1: e = leaky_relu(as[src]+ad[dst]); store; atomic max into emax[dst]
__global__ void edge_logits_kernel(const float* __restrict__ as, const float* __restrict__ ad,
                                   const int* __restrict__ ei, int E, int N, int H,
                                   float* __restrict__ ebuf, float* __restrict__ emax) {
    long long t = (long long)blockIdx.x * blockDim.x + threadIdx.x;
    long long total = (long long)(E + N) * H;
    if (t >= total) return;
    int idx = (int)(t / H);
    int hh  = (int)(t % H);
    int s, d;
    if (idx < E) { s = ei[idx]; d = ei[(long long)E + idx]; }
    else         { s = idx - E; d = s; }   // self loop
    float e = as[s * H + hh] + ad[d * H + hh];
    e = (e > 0.f) ? e : 0.2f * e;
    ebuf[t] = e;
    atomic_max_f32(&emax[d * H + hh], e);
}

// Pass 2: p = exp(e - emax[dst]); store over ebuf; atomicAdd denom[dst]
__global__ void edge_softmax_kernel(const int* __restrict__ ei, int E, int N, int H,
                                    float* __restrict__ ebuf, const float* __restrict__ emax,
                                    float* __restrict__ denom) {
    long long t = (long long)blockIdx.x * blockDim.x + threadIdx.x;
    long long total = (long long)(E + N) * H;
    if (t >= total) return;
    int idx = (int)(t / H);
    int hh  = (int)(t % H);
    int d;
    if (idx < E) d = ei[(long long)E + idx];
    else         d = idx - E;
    float p = __expf(ebuf[t] - emax[d * H + hh]);
    ebuf[t] = p;
    atomic_add_f32(&denom[d * H + hh], p);
}

// Pass 3: accum[dst,h,c] += (p/denom) * h[src,h,c]. One wave32 per edge.
__global__ __launch_bounds__(256)
void edge_aggregate_kernel(const float* __restrict__ h, const float* __restrict__ p,
                           const float* __restrict__ denom, const int* __restrict__ ei,
                           int E, int N, int H, int Cc, float* __restrict__ accum) {
    int wave = threadIdx.x >> 5, lane = threadIdx.x & 31;
    long long idx = (long long)blockIdx.x * 8 + wave;
    if (idx >= (long long)E + N) return;
    int s, d;
    if (idx < E) { s = ei[idx]; d = ei[(long long)E + idx]; }
    else         { s = (int)(idx - E); d = s; }
    int F = H * Cc;
    for (int flat = lane; flat < F; flat += 32) {
        int hh = flat / Cc;
        float w = p[idx * H + hh] / (denom[d * H + hh] + 1e-16f);
        float m = h[(long long)s * F + flat] * w;
        atomic_add_f32(&accum[(long long)d * F + flat], m);
    }
}

// In-place: a = elu(a + bias)
__global__ void bias_elu_kernel(float* __restrict__ a, const float* __restrict__ b,
                                long long n, int F) {
    long long t = (long long)blockIdx.x * blockDim.x + threadIdx.x;
    if (t >= n) return;
    float v = a[t] + b[(int)(t % F)];
    a[t] = (v > 0.f) ? v : (__expf(v) - 1.f);
}

// In-place over d_out: out[n,:] = log_softmax(out[n,:] + b2)
__global__ void logsoftmax_kernel(float* __restrict__ out, const float* __restrict__ b2, int N) {
    int n = blockIdx.x * blockDim.x + threadIdx.x;
    if (n >= N) return;
    float v[16];
    float m = -3.0e38f;
#pragma unroll
    for (int c = 0; c < 16; c++) { v[c] = out[n * 16 + c] + b2[c]; m = fmaxf(m, v[c]); }
    float s = 0.f;
#pragma unroll
    for (int c = 0; c < 16; c++) s += __expf(v[c] - m);
    float lse = __logf(s);
#pragma unroll
    for (int c = 0; c < 16; c++) out[n * 16 + c] = v[c] - m - lse;
}

static inline int cdiv_ll(long long a, long long b) { return (int)((a + b - 1) / b); }

extern "C" void kernel_launch(void* const* d_in, const int* in_sizes, int n_in,
                              void* d_out, int out_size, void* d_ws, size_t ws_size,
                              hipStream_t stream) {
    const float* x    = (const float*)d_in[0];
    const int*   ei   = (const int*)d_in[1];    // (2,E): src row then dst row
    const float* W1   = (const float*)d_in[2];  // 256 x 512
    const float* as1w = (const float*)d_in[3];  // 8 x 64
    const float* ad1w = (const float*)d_in[4];
    const float* b1   = (const float*)d_in[5];  // 512
    const float* W2   = (const float*)d_in[6];  // 512 x 16
    const float* as2w = (const float*)d_in[7];  // 1 x 16
    const float* ad2w = (const float*)d_in[8];
    const float* b2   = (const float*)d_in[9];  // 16

    const int IN_CH = 256, HID = 64, HEADS = 8, OUT_CH = 16;
    const int F1 = HEADS * HID;  // 512
    const int N = in_sizes[0] / IN_CH;
    const int E = in_sizes[1] / 2;
    const long long ET = (long long)E + N;   // edges incl. self loops

    float* ws = (float*)d_ws;
    long long off = 0;
    float* h1    = ws + off; off += (long long)N * F1;
    float* acc1  = ws + off; off += (long long)N * F1;
    float* as1   = ws + off; off += (long long)N * HEADS;
    float* ad1   = ws + off; off += (long long)N * HEADS;
    float* emax1 = ws + off; off += (long long)N * HEADS;
    float* den1  = ws + off; off += (long long)N * HEADS;
    float* eb1   = ws + off; off += ET * HEADS;
    float* h2    = ws + off; off += (long long)N * OUT_CH;
    float* as2   = ws + off; off += N;
    float* ad2   = ws + off; off += N;
    float* emax2 = ws + off; off += N;
    float* den2  = ws + off; off += N;
    float* eb2   = ws + off; off += ET;
    float* outf  = (float*)d_out;

    const float NEG_INF = -3.0e38f;

    // Init accumulators / reductions (d_out & d_ws are poisoned by harness)
    fill_kernel<<<cdiv_ll((long long)N * F1, 256), 256, 0, stream>>>(acc1, 0.f, (long long)N * F1);
    fill_kernel<<<cdiv_ll((long long)N * HEADS, 256), 256, 0, stream>>>(emax1, NEG_INF, (long long)N * HEADS);
    fill_kernel<<<cdiv_ll((long long)N * HEADS, 256), 256, 0, stream>>>(den1, 0.f, (long long)N * HEADS);
    fill_kernel<<<cdiv_ll(N, 256), 256, 0, stream>>>(emax2, NEG_INF, N);
    fill_kernel<<<cdiv_ll(N, 256), 256, 0, stream>>>(den2, 0.f, N);
    fill_kernel<<<cdiv_ll((long long)N * OUT_CH, 256), 256, 0, stream>>>(outf, 0.f, (long long)N * OUT_CH);

    // ---- Layer 1: GATConv(256 -> 64, heads=8, concat) ----
    gemm_wmma_f32<<<dim3(cdiv_ll(N, TILE_ROWS), F1 / 16), WG, 0, stream>>>(x, W1, h1, N, IN_CH, F1);
    alpha_kernel<<<cdiv_ll((long long)N * HEADS, 256), 256, 0, stream>>>(h1, as1w, ad1w, N, HEADS, HID, as1, ad1);
    edge_logits_kernel<<<cdiv_ll(ET * HEADS, 256), 256, 0, stream>>>(as1, ad1, ei, E, N, HEADS, eb1, emax1);
    edge_softmax_kernel<<<cdiv_ll(ET * HEADS, 256), 256, 0, stream>>>(ei, E, N, HEADS, eb1, emax1, den1);
    edge_aggregate_kernel<<<cdiv_ll(ET, 8), 256, 0, stream>>>(h1, eb1, den1, ei, E, N, HEADS, HID, acc1);
    bias_elu_kernel<<<cdiv_ll((long long)N * F1, 256), 256, 0, stream>>>(acc1, b1, (long long)N * F1, F1);

    // ---- Layer 2: GATConv(512 -> 16, heads=1, mean==identity) ----
    gemm_wmma_f32<<<dim3(cdiv_ll(N, TILE_ROWS), OUT_CH / 16), WG, 0, stream>>>(acc1, W2, h2, N, F1, OUT_CH);
    alpha_kernel<<<cdiv_ll(N, 256), 256, 0, stream>>>(h2, as2w, ad2w, N, 1, OUT_CH, as2, ad2);
    edge_logits_kernel<<<cdiv_ll(ET, 256), 256, 0, stream>>>(as2, ad2, ei, E, N, 1, eb2, emax2);
    edge_softmax_kernel<<<cdiv_ll(ET, 256), 256, 0, stream>>>(ei, E, N, 1, eb2, emax2, den2);
    edge_aggregate_kernel<<<cdiv_ll(ET, 8), 256, 0, stream>>>(h2, eb2, den2, ei, E, N, 1, OUT_CH, outf);
    logsoftmax_kernel<<<cdiv_ll(N, 256), 256, 0, stream>>>(outf, b2, N);
}